// FCOSBBox_8100308320993
// MI455X (gfx1250) — compile-verified
//
#include <hip/hip_runtime.h>
#include <math.h>
#include <stdint.h>

// ---------------------------------------------------------------------------
// FCOS bbox post-processing for MI455X (gfx1250, wave32).
//   Kernel A: exact two-level radix top-k per (image, level); block scan uses
//             V_WMMA_F32_16X16X4_F32 for exact integer wave sums.
//   Kernel B: TDM-staged decode + bitonic sort + early-exit greedy NMS.
// ---------------------------------------------------------------------------

#define NCLS      80
#define KTOT      4720          // 1000+1000+1000+1000+720
#define NSORT     8192
#define SCORE_THR 0.025f
#define IOU_THR   0.5f
#define IMGSZ     320.0f
#define NEG_INF   (-INFINITY)

__constant__ int   d_n[5]    = {1600, 400, 100, 25, 9};
__constant__ int   d_w[5]    = {40, 20, 10, 5, 3};
__constant__ float d_str[5]  = {8.f, 16.f, 32.f, 64.f, 128.f};
__constant__ int   d_bo[5]   = {0, 6400, 8000, 8400, 8500};   // bbox plane offsets (4*n prefix)
__constant__ int   d_co[5]   = {0, 1600, 2000, 2100, 2125};   // ctr plane offsets (n prefix)
__constant__ int   d_loff[5] = {0, 1000, 2000, 3000, 4000};   // record slot offsets per level

__device__ __forceinline__ float sigm(float x) { return 1.0f / (1.0f + expf(-x)); }

// pay = level<<24 | anchor<<7 | class
__device__ __forceinline__ unsigned packrec(int lvl, int a, int c) {
    return ((unsigned)lvl << 24) | ((unsigned)a << 7) | (unsigned)c;
}

#ifndef __has_builtin
#define __has_builtin(x) 0
#endif

// ---------------------------------------------------------------------------
// Exact wave-wide integer sum on the matrix pipe.
// D = A x B (+0) with B = all-ones: sum over column n of D equals the grand
// total of ALL 64 A inputs (independent of A's VGPR layout, since
// sum_m D[m][n] = sum_k sum_m A[m][k]).  Each lane contributes (s, 0) to A.
// C/D layout (ISA 7.12.2): lane l, reg r -> M = r + 8*(l>=16), N = l%16, so
// partial = sum of my 8 D regs; total = partial + shfl_xor(partial, 16).
// All summands are integer counts < 2^24 -> f32 accumulation is exact.
// ---------------------------------------------------------------------------
#if defined(__HIP_DEVICE_COMPILE__) && defined(__gfx1250__) && \
    __has_builtin(__builtin_amdgcn_wmma_f32_16x16x4_f32)
#define USE_WMMA_SUM 1
typedef float v2f __attribute__((ext_vector_type(2)));
typedef float v8f __attribute__((ext_vector_type(8)));
#else
#define USE_WMMA_SUM 0
#endif

__device__ __forceinline__ unsigned wave_sum_exact(unsigned s, int lane)
{
#if USE_WMMA_SUM
    v2f a; a.x = (float)s; a.y = 0.0f;     // 32 counts + 32 zeros fill A
    v2f b; b.x = 1.0f;     b.y = 1.0f;     // B = ones (layout-independent)
    v8f c = {0.f, 0.f, 0.f, 0.f, 0.f, 0.f, 0.f, 0.f};
    v8f d = __builtin_amdgcn_wmma_f32_16x16x4_f32(
        /*neg_a=*/false, a, /*neg_b=*/false, b,
        /*c_mod=*/(short)0, c, /*reuse_a=*/false, /*reuse_b=*/false);
    float p = d[0] + d[1] + d[2] + d[3] + d[4] + d[5] + d[6] + d[7];
    p += __shfl_xor(p, 16, 32);            // column sum = grand total, all lanes
    (void)lane;
    return (unsigned)p;
#else
    unsigned v = s;
#pragma unroll
    for (int dd = 16; dd > 0; dd >>= 1) v += __shfl_xor(v, dd, 32);
    (void)lane;
    return v;
#endif
}

// ---------------------------------------------------------------------------
// Block-wide threshold-bin finder over an 8192-bin histogram.
// Finds bin b with  cnt(key>b) < target <= cnt(key>=b); r = target - cnt(key>b).
// Requires blockDim.x == 1024 (32 waves). wsuf: 32-entry LDS scratch.
// ---------------------------------------------------------------------------
__device__ void find_thresh(unsigned* hist, unsigned* wsuf, unsigned target,
                            unsigned* out_b, unsigned* out_r)
{
    const int tid  = threadIdx.x;
    const int lane = tid & 31;
    const int wv   = tid >> 5;

    unsigned s = 0;
#pragma unroll
    for (int q = 0; q < 8; ++q) s += hist[tid * 8 + q];

    // per-wave total via one WMMA
    unsigned wtot = wave_sum_exact(s, lane);
    if (lane == 0) wsuf[wv] = wtot;

    // within-wave inclusive suffix scan of s (higher lanes accumulated)
    unsigned v = s;
#pragma unroll
    for (int d = 1; d < 32; d <<= 1) {
        unsigned o = __shfl_down(v, d, 32);
        if (lane + d < 32) v += o;
    }
    __syncthreads();

    if (wv == 0) {                         // suffix scan of the 32 wave totals
        unsigned y = wsuf[lane];
#pragma unroll
        for (int d = 1; d < 32; d <<= 1) {
            unsigned o = __shfl_down(y, d, 32);
            if (lane + d < 32) y += o;
        }
        wsuf[lane] = y;                    // inclusive: sum of waves >= lane
    }
    __syncthreads();

    unsigned above_waves = (wv < 31) ? wsuf[wv + 1] : 0u;
    unsigned running = above_waves + (v - s);   // strictly above my 8 bins
    for (int q = 7; q >= 0; --q) {
        unsigned b = (unsigned)tid * 8 + (unsigned)q;
        unsigned h = hist[b];
        if (running < target && target <= running + h) { *out_b = b; *out_r = target - running; }
        running += h;
    }
    __syncthreads();
}

// ---------------------------------------------------------------------------
// Kernel A: per-(image,level) exact top-k selection.
//   key = valid ? float_bits(sigmoid(logit)) : 0   (monotonic, sigmoid > 0)
//   pass1: 13-bit histogram -> threshold bin b1
//   pass2: next-13-bit histogram within b1 -> b2
//   pass3: emit bin>b1, emit b1&&bin2>b2, exact rank-select inside (b1,b2)
// ---------------------------------------------------------------------------
__global__ __launch_bounds__(1024, 1)
void fcos_select_topk(const float* __restrict__ cls0, const float* __restrict__ cls1,
                      const float* __restrict__ cls2, const float* __restrict__ cls3,
                      const float* __restrict__ cls4,
                      float* __restrict__ rec_sc, unsigned* __restrict__ rec_pay,
                      unsigned* __restrict__ counters)
{
    const int img = blockIdx.x / 5, lvl = blockIdx.x % 5;
    const int tid = threadIdx.x;
    const int n = d_n[lvl];
    const float* cls;
    switch (lvl) {
        case 0: cls = cls0; break; case 1: cls = cls1; break;
        case 2: cls = cls2; break; case 3: cls = cls3; break;
        default: cls = cls4; break;
    }
    const float* p = cls + (size_t)img * (size_t)(n * NCLS);
    const unsigned base = (unsigned)img * KTOT + (unsigned)d_loff[lvl];

    if (lvl == 4) {  // k == M: take everything, no selection needed
        for (int c = 0; c < NCLS; ++c)
            for (int a = tid; a < n; a += 1024) {
                float sc = sigm(p[c * n + a]);
                unsigned slot = base + (unsigned)(c * n + a);
                rec_sc[slot]  = sc;
                rec_pay[slot] = packrec(4, a, c);
            }
        return;
    }

    __shared__ unsigned hist[8192];
    __shared__ unsigned wsuf[32];
    __shared__ unsigned sidek[512];
    __shared__ unsigned sidep[512];
    __shared__ unsigned s_b1, s_r1, s_b2, s_r2, s_scnt;
    const unsigned k = 1000u;

    // ---- pass 1: coarse histogram on key[31:19]
    for (int b = tid; b < 8192; b += 1024) hist[b] = 0u;
    __syncthreads();
    for (int c = 0; c < NCLS; ++c) {
        const float* pc = p + c * n;
        for (int a = tid; a < n; a += 1024) {
            float sc = sigm(pc[a]);
            unsigned key = (sc > SCORE_THR) ? __float_as_uint(sc) : 0u;
            atomicAdd(&hist[key >> 19], 1u);
        }
    }
    __syncthreads();
    find_thresh(hist, wsuf, k, &s_b1, &s_r1);
    const unsigned b1 = s_b1, r1 = s_r1;

    // ---- pass 2: refine within threshold bin on the next 13 bits
    for (int b = tid; b < 8192; b += 1024) hist[b] = 0u;
    __syncthreads();
    for (int c = 0; c < NCLS; ++c) {
        const float* pc = p + c * n;
        for (int a = tid; a < n; a += 1024) {
            float sc = sigm(pc[a]);
            unsigned key = (sc > SCORE_THR) ? __float_as_uint(sc) : 0u;
            if ((key >> 19) == b1) atomicAdd(&hist[(key >> 6) & 8191u], 1u);
        }
    }
    __syncthreads();
    find_thresh(hist, wsuf, r1, &s_b2, &s_r2);
    const unsigned b2 = s_b2, r2 = s_r2;

    // ---- pass 3: emission + exact tie resolution in (b1,b2)
    if (tid == 0) s_scnt = 0u;
    __syncthreads();
    for (int c = 0; c < NCLS; ++c) {
        const float* pc = p + c * n;
        for (int a = tid; a < n; a += 1024) {
            float sc = sigm(pc[a]);
            unsigned key = (sc > SCORE_THR) ? __float_as_uint(sc) : 0u;
            unsigned bin = key >> 19;
            bool emit = false;
            if (bin > b1) emit = true;
            else if (bin == b1) {
                unsigned bin2 = (key >> 6) & 8191u;
                if (bin2 > b2) emit = true;
                else if (bin2 == b2) {
                    unsigned pos = atomicAdd(&s_scnt, 1u);
                    if (pos < 512u) { sidek[pos] = key; sidep[pos] = packrec(lvl, a, c); }
                }
            }
            if (emit) {
                unsigned pos = atomicAdd(&counters[blockIdx.x], 1u);
                if (pos < k) { rec_sc[base + pos] = sc; rec_pay[base + pos] = packrec(lvl, a, c); }
            }
        }
    }
    __syncthreads();
    unsigned cnt = s_scnt; if (cnt > 512u) cnt = 512u;
    for (unsigned e = tid; e < cnt; e += 1024) {   // O(cnt^2) rank select, cnt is tiny
        unsigned ke = sidek[e];
        unsigned rank = 0;
        for (unsigned j = 0; j < cnt; ++j) {
            unsigned kj = sidek[j];
            rank += (kj > ke) || (kj == ke && j < e);
        }
        if (rank < r2) {
            unsigned pos = atomicAdd(&counters[blockIdx.x], 1u);
            if (pos < k) { rec_sc[base + pos] = __uint_as_float(ke); rec_pay[base + pos] = sidep[e]; }
        }
    }
}

// ---------------------------------------------------------------------------
// TDM (async tensor) staging: gfx1250-specific path with toolchain dispatch.
// ---------------------------------------------------------------------------
#if defined(__HIP_DEVICE_COMPILE__) && defined(__gfx1250__) && \
    __has_builtin(__builtin_amdgcn_tensor_load_to_lds) && \
    __has_builtin(__builtin_amdgcn_s_wait_tensorcnt)
#define USE_TDM 1
#if defined(__has_include)
#if __has_include(<hip/amd_detail/amd_gfx1250_TDM.h>)
#define TDM_ARGS6 1
#endif
#endif
#else
#define USE_TDM 0
#endif

#if USE_TDM
typedef unsigned tdm_u32x4 __attribute__((ext_vector_type(4)));
typedef int      tdm_i32x4 __attribute__((ext_vector_type(4)));
typedef int      tdm_i32x8 __attribute__((ext_vector_type(8)));

// 1-D copy of ndw dwords, expressed as a 2-D (tile_dim0=ndw, tile_dim1=1) tile
// per CDNA5 ISA ch8 D# layout. Issued once per wave (TDM ignores EXEC).
__device__ __forceinline__ void tdm_load_1d(const float* gsrc, const float* ldst, unsigned ndw)
{
    unsigned long long ga = (unsigned long long)(uintptr_t)gsrc;
    unsigned la = (unsigned)(uintptr_t)ldst;   // flat addr low 32 bits == LDS byte offset
    tdm_u32x4 g0;
    g0.x = 1u;                                             // count=1, user descriptor
    g0.y = la;                                             // lds_addr
    g0.z = (unsigned)ga;                                   // global_addr[31:0]
    g0.w = (unsigned)((ga >> 32) & 0x01FFFFFFull) | (2u << 30);  // addr[56:32] | type=2
    tdm_i32x8 g1;
    g1.s0 = (int)(2u << 16);                               // data_size=4B, mask=0
    g1.s1 = (int)((ndw & 0xFFFFu) << 16);                  // tensor_dim0[15:0] @ bits 63:48
    g1.s2 = (int)(((ndw >> 16) & 0xFFFFu) | (1u << 16));   // tensor_dim0[31:16] | tensor_dim1=1
    g1.s3 = (int)((ndw & 0xFFFFu) << 16);                  // tensor_dim1 hi=0 | tile_dim0=ndw
    g1.s4 = 1;                                             // tile_dim1=1, tile_dim2=0
    g1.s5 = (int)ndw;                                      // tensor_dim0_stride[31:0]
    g1.s6 = 0;                                             // stride hi | dim1_stride lo
    g1.s7 = 0;
    tdm_i32x4 z4; z4.x = z4.y = z4.z = z4.w = 0;
#if defined(TDM_ARGS6)
    tdm_i32x8 z8; z8.s0=z8.s1=z8.s2=z8.s3=z8.s4=z8.s5=z8.s6=z8.s7=0;
    __builtin_amdgcn_tensor_load_to_lds(g0, g1, z4, z4, z8, 0);
#else
    __builtin_amdgcn_tensor_load_to_lds(g0, g1, z4, z4, 0);
#endif
}
#endif // USE_TDM

// decode candidate -> clipped box + label, reading LDS-staged bbox planes
__device__ __forceinline__ void decode_cand(unsigned pay, const float* s_bbox,
                                            float& x1, float& y1, float& x2, float& y2, int& lab)
{
    int lvl = (int)(pay >> 24);
    int a   = (int)((pay >> 7) & 0x1FFFFu);
    lab     = (int)(pay & 0x7Fu);
    int n = d_n[lvl], w = d_w[lvl], bo = d_bo[lvl];
    float st = d_str[lvl];
    int ax = a % w, ay = a / w;
    float px = ((float)ax + 0.5f) * st;
    float py = ((float)ay + 0.5f) * st;
    float l = s_bbox[bo + a];
    float t = s_bbox[bo + n + a];
    float r = s_bbox[bo + 2 * n + a];
    float b = s_bbox[bo + 3 * n + a];
    x1 = fminf(fmaxf(px - l, 0.f), IMGSZ);
    y1 = fminf(fmaxf(py - t, 0.f), IMGSZ);
    x2 = fminf(fmaxf(px + r, 0.f), IMGSZ);
    y2 = fminf(fmaxf(py + b, 0.f), IMGSZ);
}

// ---------------------------------------------------------------------------
// Kernel B: one block per image. TDM-stage planes, build final keys,
// bitonic sort 8192 desc, wave32 greedy NMS with early exit at 100 kept.
// ---------------------------------------------------------------------------
__global__ __launch_bounds__(1024, 1)
void fcos_decode_sort_nms(const float* __restrict__ bb0, const float* __restrict__ bb1,
                          const float* __restrict__ bb2, const float* __restrict__ bb3,
                          const float* __restrict__ bb4,
                          const float* __restrict__ ct0, const float* __restrict__ ct1,
                          const float* __restrict__ ct2, const float* __restrict__ ct3,
                          const float* __restrict__ ct4,
                          const float* __restrict__ rec_sc, const unsigned* __restrict__ rec_pay,
                          float* __restrict__ out)
{
    const int img = blockIdx.x;
    const int tid = threadIdx.x;

    __shared__ float    s_bbox[8536];   // all levels, [4][n] per level
    __shared__ float    s_ctr[2134];
    __shared__ float    s_key[NSORT];
    __shared__ unsigned s_pay[NSORT];
    __shared__ int      s_nk;

#if USE_TDM
    if (tid < 32) {  // wave 0 drives the Tensor Data Mover
        tdm_load_1d(bb0 + (size_t)img * 6400, s_bbox + 0,    6400);
        tdm_load_1d(bb1 + (size_t)img * 1600, s_bbox + 6400, 1600);
        tdm_load_1d(bb2 + (size_t)img * 400,  s_bbox + 8000, 400);
        tdm_load_1d(bb3 + (size_t)img * 100,  s_bbox + 8400, 100);
        tdm_load_1d(bb4 + (size_t)img * 36,   s_bbox + 8500, 36);
        tdm_load_1d(ct0 + (size_t)img * 1600, s_ctr + 0,     1600);
        tdm_load_1d(ct1 + (size_t)img * 400,  s_ctr + 1600,  400);
        tdm_load_1d(ct2 + (size_t)img * 100,  s_ctr + 2000,  100);
        tdm_load_1d(ct3 + (size_t)img * 25,   s_ctr + 2100,  25);
        tdm_load_1d(ct4 + (size_t)img * 9,    s_ctr + 2125,  9);
        __builtin_amdgcn_s_wait_tensorcnt(0);
    }
#else
    for (int i = tid; i < 6400; i += 1024) s_bbox[i]        = bb0[(size_t)img * 6400 + i];
    for (int i = tid; i < 1600; i += 1024) s_bbox[6400 + i] = bb1[(size_t)img * 1600 + i];
    for (int i = tid; i < 400;  i += 1024) s_bbox[8000 + i] = bb2[(size_t)img * 400 + i];
    for (int i = tid; i < 100;  i += 1024) s_bbox[8400 + i] = bb3[(size_t)img * 100 + i];
    for (int i = tid; i < 36;   i += 1024) s_bbox[8500 + i] = bb4[(size_t)img * 36 + i];
    for (int i = tid; i < 1600; i += 1024) s_ctr[i]         = ct0[(size_t)img * 1600 + i];
    for (int i = tid; i < 400;  i += 1024) s_ctr[1600 + i]  = ct1[(size_t)img * 400 + i];
    for (int i = tid; i < 100;  i += 1024) s_ctr[2000 + i]  = ct2[(size_t)img * 100 + i];
    for (int i = tid; i < 25;   i += 1024) s_ctr[2100 + i]  = ct3[(size_t)img * 25 + i];
    for (int i = tid; i < 9;    i += 1024) s_ctr[2125 + i]  = ct4[(size_t)img * 9 + i];
#endif
    __syncthreads();

    // ---- build sort keys: final = valid ? sc * sigmoid(ctr) : -inf
    for (int j = tid; j < NSORT; j += 1024) {
        if (j < KTOT) {
            float sc = rec_sc[(size_t)img * KTOT + j];
            unsigned pay = rec_pay[(size_t)img * KTOT + j];
            int lvl = (int)(pay >> 24);
            int a   = (int)((pay >> 7) & 0x1FFFFu);
            float sf = sigm(s_ctr[d_co[lvl] + a]);
            s_key[j] = (sc > SCORE_THR) ? sc * sf : NEG_INF;
            s_pay[j] = pay;
        } else { s_key[j] = NEG_INF; s_pay[j] = 0u; }
    }

    // ---- bitonic sort, descending by key, deterministic payload tiebreak
    for (unsigned kk = 2; kk <= NSORT; kk <<= 1) {
        for (unsigned jj = kk >> 1; jj > 0; jj >>= 1) {
            __syncthreads();
            for (unsigned i = tid; i < NSORT; i += 1024) {
                unsigned ixj = i ^ jj;
                if (ixj > i) {
                    float    ki = s_key[i], kj = s_key[ixj];
                    unsigned pi = s_pay[i], pj = s_pay[ixj];
                    bool iBefore = (ki > kj) || (ki == kj && pi < pj);
                    bool jBefore = (kj > ki) || (kj == ki && pj < pi);
                    bool dirDesc = ((i & kk) == 0u);
                    bool sw = dirDesc ? jBefore : iBefore;
                    if (sw) { s_key[i] = kj; s_key[ixj] = ki; s_pay[i] = pj; s_pay[ixj] = pi; }
                }
            }
        }
    }
    __syncthreads();

    // ---- greedy NMS on wave 0; kept list lives in registers across lanes.
    // Per-class offset trick == "same label" check; first 100 kept are the
    // output (candidates arrive in descending final order), so exit early.
    if (tid < 32) {
        const int lane = tid;
        float kx1[4], ky1[4], kx2[4], ky2[4]; int klab[4];
        int nk = 0;
        for (int j = 0; j < KTOT; ++j) {
            float fin = s_key[j];
            if (fin == NEG_INF) break;        // sorted: the rest are invalid
            unsigned pay = s_pay[j];
            float x1, y1, x2, y2; int lab;
            decode_cand(pay, s_bbox, x1, y1, x2, y2, lab);
            float areaC = fmaxf(x2 - x1, 0.f) * fmaxf(y2 - y1, 0.f);
            bool sup = false;
#pragma unroll
            for (int q = 0; q < 4; ++q) {
                int idx = q * 32 + lane;
                if (idx < nk && klab[q] == lab) {
                    float ix1 = fmaxf(x1, kx1[q]), iy1 = fmaxf(y1, ky1[q]);
                    float ix2 = fminf(x2, kx2[q]), iy2 = fminf(y2, ky2[q]);
                    float inter = fmaxf(ix2 - ix1, 0.f) * fmaxf(iy2 - iy1, 0.f);
                    float areaK = fmaxf(kx2[q] - kx1[q], 0.f) * fmaxf(ky2[q] - ky1[q], 0.f);
                    float iou = inter / fmaxf(areaC + areaK - inter, 1e-6f);
                    sup = sup || (iou > IOU_THR);
                }
            }
            if (__ballot((int)sup) == 0ull) {
                int q = nk >> 5, owner = nk & 31;
                if (lane == owner) { kx1[q] = x1; ky1[q] = y1; kx2[q] = x2; ky2[q] = y2; klab[q] = lab; }
                if (lane == 0) {
                    size_t ro = (size_t)img * 500 + (size_t)nk * 5;
                    out[ro + 0] = x1; out[ro + 1] = y1; out[ro + 2] = x2; out[ro + 3] = y2;
                    out[ro + 4] = fin;
                    out[64000 + img * 100 + nk] = (float)lab;
                }
                ++nk;
                if (nk == 100) break;
            }
        }
        if (lane == 0) s_nk = nk;
    }
    __syncthreads();

    const int nk = s_nk;
    for (int r = tid; r < 100; r += 1024) {
        if (r >= nk) {
            size_t ro = (size_t)img * 500 + (size_t)r * 5;
            out[ro + 0] = 0.f; out[ro + 1] = 0.f; out[ro + 2] = 0.f;
            out[ro + 3] = 0.f; out[ro + 4] = 0.f;
            out[64000 + img * 100 + r] = -1.0f;
        }
    }
}

// ---------------------------------------------------------------------------
// Host launcher. Workspace: [0,4096) counters, then rec_sc[128*4720],
// then rec_pay[128*4720]  (~4.9 MB total, zeroed every call -> deterministic).
// ---------------------------------------------------------------------------
extern "C" void kernel_launch(void* const* d_in, const int* in_sizes, int n_in,
                              void* d_out, int out_size, void* d_ws, size_t ws_size,
                              hipStream_t stream)
{
    (void)in_sizes; (void)n_in; (void)out_size; (void)ws_size;
    const float* cls[5]; const float* bb[5]; const float* ct[5];
    for (int i = 0; i < 5; ++i) {
        cls[i] = (const float*)d_in[3 * i + 0];
        bb[i]  = (const float*)d_in[3 * i + 1];
        ct[i]  = (const float*)d_in[3 * i + 2];
    }
    unsigned* counters = (unsigned*)d_ws;
    float*    rec_sc   = (float*)((char*)d_ws + 4096);
    unsigned* rec_pay  = (unsigned*)((char*)d_ws + 4096 + (size_t)128 * KTOT * 4);
    size_t clear_bytes = 4096 + (size_t)128 * KTOT * 8;
    hipMemsetAsync(d_ws, 0, clear_bytes, stream);

    fcos_select_topk<<<640, 1024, 0, stream>>>(cls[0], cls[1], cls[2], cls[3], cls[4],
                                               rec_sc, rec_pay, counters);
    fcos_decode_sort_nms<<<128, 1024, 0, stream>>>(bb[0], bb[1], bb[2], bb[3], bb[4],
                                                   ct[0], ct[1], ct[2], ct[3], ct[4],
                                                   rec_sc, rec_pay, (float*)d_out);
}